// NodeBlock_29119878266987
// MI455X (gfx1250) — compile-verified
//
#include <hip/hip_runtime.h>

typedef __attribute__((ext_vector_type(2))) float v2f;
typedef __attribute__((ext_vector_type(8))) float v8f;

#define D_MODEL   128
#define D_IN      256
#define N_NODES_  50000
#define N_EDGES_  800000
#define LDS_STRIDE 258   // 256 + 2 pad: kills 16-way LDS bank conflict on A-tile rows

// ---------------------------------------------------------------------------
// Kernel 0: zero the workspace (seg_sum [N_NODES*128] + counts [N_NODES])
// ---------------------------------------------------------------------------
__global__ __launch_bounds__(256) void zero_ws_kernel(float4* __restrict__ ws, int n4) {
    int i = blockIdx.x * 256 + threadIdx.x;
    if (i < n4) {
        float4 z; z.x = 0.f; z.y = 0.f; z.z = 0.f; z.w = 0.f;
        ws[i] = z;
    }
}

// ---------------------------------------------------------------------------
// Kernel 1: scatter-add edges into seg_sum / counts.
// One wave32 per edge row: lane l handles columns 4l..4l+3 via float4 load
// (512B per row, fully coalesced) + 4 hardware f32 atomics (L2-resident dest).
// ---------------------------------------------------------------------------
__global__ __launch_bounds__(256) void scatter_kernel(const float* __restrict__ edges,
                                                      const int*   __restrict__ recv,
                                                      float*       __restrict__ seg,
                                                      float*       __restrict__ cnt,
                                                      int n_edges) {
    int gtid = blockIdx.x * 256 + threadIdx.x;
    int e    = gtid >> 5;
    int lane = gtid & 31;
    if (e >= n_edges) return;

    int r = recv[e];
    float4 v = ((const float4*)edges)[(size_t)e * 32 + lane];
    float* dst = seg + (size_t)r * D_MODEL + lane * 4;
    unsafeAtomicAdd(dst + 0, v.x);
    unsafeAtomicAdd(dst + 1, v.y);
    unsafeAtomicAdd(dst + 2, v.z);
    unsafeAtomicAdd(dst + 3, v.w);
    if (lane == 0) unsafeAtomicAdd(cnt + r, 1.0f);
}

// ---------------------------------------------------------------------------
// Kernel 2: agg = seg/max(cnt,1); collected = [agg | nodes]; out = collected @ W
// Block = one 16-row M tile (3125 blocks, 50000 = 3125*16 exactly -> no tail,
// EXEC is all-1s for every WMMA). 8 waves per block, one 16-col N tile each.
// K = 256 swept with 64 x v_wmma_f32_16x16x4_f32 per wave (exact f32 math).
// ---------------------------------------------------------------------------
__global__ __launch_bounds__(256) void nodeblock_gemm_kernel(const float* __restrict__ seg,
                                                             const float* __restrict__ cnt,
                                                             const float* __restrict__ nodes,
                                                             const float* __restrict__ W,
                                                             float*       __restrict__ out) {
    __shared__ __align__(16) float ldsA[16 * LDS_STRIDE];
    __shared__ float ldsInv[16];

    const int tid   = threadIdx.x;
    const int tileM = blockIdx.x * 16;

    if (tid < 16) {
        float c = cnt[tileM + tid];
        ldsInv[tid] = 1.0f / fmaxf(c, 1.0f);
    }
    __syncthreads();

    // Stage collected tile A[16][256] into LDS, fusing the segment-mean divide.
    for (int i = tid; i < 16 * 256; i += 256) {
        int row = i >> 8;
        int col = i & 255;
        int m   = tileM + row;
        float v;
        if (col < D_MODEL) v = seg[(size_t)m * D_MODEL + col] * ldsInv[row];
        else               v = nodes[(size_t)m * D_MODEL + (col - D_MODEL)];
        ldsA[row * LDS_STRIDE + col] = v;
    }
    __syncthreads();

    const int wave    = tid >> 5;       // 0..7 -> which 16-wide N tile
    const int lane    = tid & 31;
    const int nl      = lane & 15;      // N (for B/D) and M (for A) within tile
    const int g       = lane >> 4;      // K sub-group: lanes 16..31 hold K+2
    const int n0      = wave * 16;

    const float* wcol = W + n0 + nl;                 // column nl of wave's W tile
    const float* arow = &ldsA[nl * LDS_STRIDE];      // row (M = lane&15) of A tile

    v8f c = {};
#pragma unroll 8
    for (int k0 = 0; k0 < D_IN; k0 += 4) {
        int k = k0 + 2 * g;
        // A 16x4 layout: lanes 0-15 K={k0,k0+1}, lanes 16-31 K={k0+2,k0+3}
        v2f a = *(const v2f*)(&arow[k]);
        // B 4x16 layout: VGPR0 K=k0(+2g), VGPR1 K=k0+1(+2g), N = lane&15
        v2f b;
        b.x = wcol[(size_t)k * D_MODEL];
        b.y = wcol[(size_t)(k + 1) * D_MODEL];
        // 8 args: (neg_a, A, neg_b, B, c_mod, C, reuse_a, reuse_b)
        c = __builtin_amdgcn_wmma_f32_16x16x4_f32(false, a, false, b,
                                                  (short)0, c, false, false);
    }

    // D layout: VGPR r -> row (r + 8*g), col = lane&15
    float* obase = out + (size_t)(tileM + g * 8) * D_MODEL + n0 + nl;
#pragma unroll
    for (int r = 0; r < 8; ++r) {
        obase[(size_t)r * D_MODEL] = c[r];
    }
}

// ---------------------------------------------------------------------------
extern "C" void kernel_launch(void* const* d_in, const int* in_sizes, int n_in,
                              void* d_out, int out_size, void* d_ws, size_t ws_size,
                              hipStream_t stream) {
    const float* edges = (const float*)d_in[0];   // [N_EDGES, 128] f32
    const float* nodes = (const float*)d_in[1];   // [N_NODES, 128] f32
    const int*   recv  = (const int*)  d_in[2];   // [N_EDGES] receivers
    const float* W     = (const float*)d_in[3];   // [256, 128] f32
    float*       out   = (float*)d_out;           // [N_NODES, 128] f32

    const int n_edges = in_sizes[0] / D_MODEL;    // 800000
    const int n_nodes = in_sizes[1] / D_MODEL;    // 50000

    float* seg = (float*)d_ws;                    // [n_nodes * 128]
    float* cnt = seg + (size_t)n_nodes * D_MODEL; // [n_nodes]

    const int n4 = (n_nodes * D_MODEL + n_nodes) / 4;   // 6,450,000 / 4
    zero_ws_kernel<<<(n4 + 255) / 256, 256, 0, stream>>>((float4*)d_ws, n4);

    const int scatter_threads = n_edges * 32;
    scatter_kernel<<<(scatter_threads + 255) / 256, 256, 0, stream>>>(
        edges, recv, seg, cnt, n_edges);

    nodeblock_gemm_kernel<<<n_nodes / 16, 256, 0, stream>>>(
        seg, cnt, nodes, W, out);
}